// SAGEConv_87677462380712
// MI455X (gfx1250) — compile-verified
//
#include <hip/hip_runtime.h>
#include <hip/hip_bf16.h>

typedef __attribute__((ext_vector_type(2))) float v2f;
typedef __attribute__((ext_vector_type(8))) float v8f;

#define D_FEAT 128

// ---------------------------------------------------------------------------
// Kernel 1: zero the workspace accumulators (summed [N*128] and deg [N]).
// Grid-stride; graph-capture safe (no hipMemset).
// ---------------------------------------------------------------------------
__global__ void sage_zero_ws(float* __restrict__ p, long n) {
    long i = (long)blockIdx.x * blockDim.x + threadIdx.x;
    long stride = (long)gridDim.x * blockDim.x;
    for (; i < n; i += stride) p[i] = 0.0f;
}

// ---------------------------------------------------------------------------
// Kernel 2: edge-parallel scatter-add. 32 lanes per edge; each lane moves a
// float4 (128 floats / 32 lanes = 4 per lane). summed[] is L2-resident
// (51.2 MB < 192 MB L2), so the f32 atomics resolve in the L2 atomic units.
// ---------------------------------------------------------------------------
__global__ __launch_bounds__(256) void sage_scatter(
    const float* __restrict__ feat, const int* __restrict__ src,
    const int* __restrict__ dst, float* __restrict__ summed,
    float* __restrict__ deg, int n_edges)
{
    long gid = (long)blockIdx.x * blockDim.x + threadIdx.x;
    int e = (int)(gid >> 5);
    int lane = (int)(gid & 31);
    if (e >= n_edges) return;
    int s = src[e];
    int d = dst[e];
    const float4 v = *(const float4*)(feat + (long)s * D_FEAT + lane * 4);
    float* acc = summed + (long)d * D_FEAT + lane * 4;
    atomicAdd(acc + 0, v.x);
    atomicAdd(acc + 1, v.y);
    atomicAdd(acc + 2, v.z);
    atomicAdd(acc + 3, v.w);
    if (lane == 0) atomicAdd(deg + d, 1.0f);
}

// ---------------------------------------------------------------------------
// Kernel 3: h_neigh = summed / max(deg, 1)   (in place)
// ---------------------------------------------------------------------------
__global__ __launch_bounds__(256) void sage_norm(
    float* __restrict__ summed, const float* __restrict__ deg, long n_elts)
{
    long i = (long)blockIdx.x * blockDim.x + threadIdx.x;
    if (i >= n_elts) return;
    float dg = deg[i >> 7];                 // element / 128 -> node
    float inv = dg > 1.0f ? 1.0f / dg : 1.0f;
    summed[i] *= inv;
}

// ---------------------------------------------------------------------------
// Kernel 4: fused dense part with native fp32 WMMA:
//   out = feat @ W_self^T + h_neigh @ W_neigh^T
// One block = one 16-node row tile; 8 waves = 8 column tiles of 16.
// Per wave: 32 K-steps x 2 x v_wmma_f32_16x16x4_f32 into one v8f accumulator.
//
// A (16x4 f32) layout: lane l<16 holds A[l][k0],A[l][k0+1]; lane l+16 holds
// A[l][k0+2],A[l][k0+3]  -> one float2 load at column k0 + 2*(lane>=16).
// B (4x16 f32) with B[k][n] = W[n][k] (W is [out,in] row-major) -> identical
// float2 addressing on W rows.
// C/D (16x16 f32, 8 VGPRs): VGPR r = row M=r (lanes 0-15) / M=r+8 (16-31).
// ---------------------------------------------------------------------------
__global__ __launch_bounds__(256) void sage_gemm_wmma(
    const float* __restrict__ feat, const float* __restrict__ hneigh,
    const float* __restrict__ Wself, const float* __restrict__ Wneigh,
    float* __restrict__ out)
{
    const int wave = threadIdx.x >> 5;          // 0..7 -> output column tile
    const int lane = threadIdx.x & 31;
    const int m = lane & 15;                    // row-within-tile / N index
    const int khalf = (lane >> 4) * 2;          // 0 or 2: K sub-column
    const long node_base = (long)blockIdx.x * 16;
    const int  col_base = wave * 16;

    const float* fa = feat   + (node_base + m) * D_FEAT + khalf;
    const float* ha = hneigh + (node_base + m) * D_FEAT + khalf;
    const float* ws = Wself  + (long)(col_base + m) * D_FEAT + khalf;
    const float* wn = Wneigh + (long)(col_base + m) * D_FEAT + khalf;

    v8f c = {};
#pragma unroll
    for (int k0 = 0; k0 < D_FEAT; k0 += 4) {
        v2f a  = *(const v2f*)(fa + k0);
        v2f bs = *(const v2f*)(ws + k0);
        c = __builtin_amdgcn_wmma_f32_16x16x4_f32(
                /*neg_a=*/false, a, /*neg_b=*/false, bs,
                /*c_mod=*/(short)0, c, /*reuse_a=*/false, /*reuse_b=*/false);
        v2f ah = *(const v2f*)(ha + k0);
        v2f bn = *(const v2f*)(wn + k0);
        c = __builtin_amdgcn_wmma_f32_16x16x4_f32(
                false, ah, false, bn, (short)0, c, false, false);
    }

    // D tile store: row = node_base + r + 8*(lane>=16), col = col_base + m
    float* o = out + (node_base + (lane >> 4) * 8) * D_FEAT + col_base + m;
#pragma unroll
    for (int r = 0; r < 8; ++r)
        o[(long)r * D_FEAT] = c[r];
}

// ---------------------------------------------------------------------------
extern "C" void kernel_launch(void* const* d_in, const int* in_sizes, int n_in,
                              void* d_out, int out_size, void* d_ws, size_t ws_size,
                              hipStream_t stream)
{
    const float* feat   = (const float*)d_in[0];   // [N, 128]
    const int*   src    = (const int*)d_in[1];     // [E]
    const int*   dst    = (const int*)d_in[2];     // [E]
    const float* Wself  = (const float*)d_in[3];   // [128, 128] row-major [out,in]
    const float* Wneigh = (const float*)d_in[4];   // [128, 128]
    float*       out    = (float*)d_out;           // [N, 128]

    const int n_nodes = in_sizes[0] / D_FEAT;      // 100000
    const int n_edges = in_sizes[1];               // 1600000

    // Workspace layout: summed [N*128] then deg [N]  (~51.6 MB)
    float* summed = (float*)d_ws;
    float* deg    = summed + (long)n_nodes * D_FEAT;

    const long n_acc = (long)n_nodes * D_FEAT + n_nodes;
    sage_zero_ws<<<4096, 256, 0, stream>>>(summed, n_acc);

    const long scatter_threads = (long)n_edges * 32;
    const int  scatter_blocks  = (int)((scatter_threads + 255) / 256);
    sage_scatter<<<scatter_blocks, 256, 0, stream>>>(feat, src, dst, summed, deg, n_edges);

    const long n_elts = (long)n_nodes * D_FEAT;
    sage_norm<<<(int)((n_elts + 255) / 256), 256, 0, stream>>>(summed, deg, n_elts);

    // 100000 % 16 == 0 -> all tiles full, EXEC stays all-ones through WMMA.
    sage_gemm_wmma<<<n_nodes / 16, 256, 0, stream>>>(feat, summed, Wself, Wneigh, out);
}